// MeshLoss_28432683500145
// MI455X (gfx1250) — compile-verified
//
#include <hip/hip_runtime.h>

// ---------------- problem constants ----------------
#define NB      2        // batches
#define MESH_N  32       // coarse mesh is 32x32
#define FINE_N  94       // (32-1)*3 + 1
#define NV      (FINE_N*FINE_N)   // 8836 vertices per batch
#define NTILES  553               // ceil(8836/16)
#define NPAD    (NTILES*16)       // 8848
#define NP      16384             // points per batch
#define MTILES  (NP/16)           // 1024 point tiles per batch
#define WAVES_PER_BLOCK 8
#define CHUNK_TILES 64            // LDS staging: 64 tiles * 32 lanes * 8B = 16KB

typedef float v2f __attribute__((ext_vector_type(2)));
typedef float v8f __attribute__((ext_vector_type(8)));

// mesh(b,c,x,y) = vertices[b,c,x,3,y]  with shape (2,3,32,4,32)
__device__ __forceinline__ float mesh_at(const float* V, int b, int c, int x, int y) {
    return V[(b * 3 + c) * 4096 + x * 128 + 96 + y];
}

// coarse(b,c,x,j): y-refined mesh, j in [0,94)
__device__ __forceinline__ float coarse_at(const float* V, int b, int c, int x, int j) {
    if (j == FINE_N - 1) {  // appended last column: zero except x==31 -> mesh[31,31]
        return (x == MESH_N - 1) ? mesh_at(V, b, c, MESH_N - 1, MESH_N - 1) : 0.0f;
    }
    int jc = j / 3, s = j % 3;
    float w = (float)s * (1.0f / 3.0f);
    return mesh_at(V, b, c, x, jc) * (1.0f - w) + mesh_at(V, b, c, x, jc + 1) * w;
}

// fine(b,c,i,j): fully refined (94x94)
__device__ __forceinline__ float fine_at(const float* V, int b, int c, int i, int j) {
    if (i == FINE_N - 1) return coarse_at(V, b, c, MESH_N - 1, j);
    int ic = i / 3, r = i % 3;
    float w = (float)r * (1.0f / 3.0f);
    return coarse_at(V, b, c, ic, j) * (1.0f - w) + coarse_at(V, b, c, ic + 1, j) * w;
}

// ---------------- kernel 1: prepare packed, pre-swizzled matrices ----------------
// Bsw[b][tile][lane] (float2, WMMA B-fragment lane order):
//   lane<16 : (-2vx, -2vy)  of vertex tile*16+lane
//   lane>=16: (-2vz, |v|^2) of vertex tile*16+lane-16   (padding: (0,1e30))
// Pts[b*NP+n] = (px, py, pz, valid?1:0)
__global__ __launch_bounds__(256)
void prep_kernel(const float* __restrict__ verts, const float* __restrict__ pc,
                 float2* __restrict__ Bsw, float4* __restrict__ Pts) {
    int idx = blockIdx.x * blockDim.x + threadIdx.x;
    const int nBsw = NB * NTILES * 32;
    if (idx < nBsw) {
        int b    = idx / (NTILES * 32);
        int rem  = idx % (NTILES * 32);
        int tile = rem >> 5;
        int lane = rem & 31;
        int n    = tile * 16 + (lane & 15);
        float2 o;
        if (n < NV) {
            int i = n / FINE_N, j = n % FINE_N;
            float x = fine_at(verts, b, 0, i, j);
            float y = fine_at(verts, b, 1, i, j);
            float z = fine_at(verts, b, 2, i, j);
            if (lane < 16) o = make_float2(-2.0f * x, -2.0f * y);
            else           o = make_float2(-2.0f * z, x * x + y * y + z * z);
        } else {
            o = (lane < 16) ? make_float2(0.0f, 0.0f)
                            : make_float2(0.0f, 1.0e30f);  // padding: never the min
        }
        Bsw[idx] = o;
    } else if (idx < nBsw + NB * NP) {
        int k = idx - nBsw;
        int b = k / NP, n = k % NP;
        const float* base = pc + (size_t)b * 3 * NP;
        float x = base[n], y = base[NP + n], z = base[2 * NP + n];
        float valid = (x != 0.0f || y != 0.0f || z != 0.0f) ? 1.0f : 0.0f;
        Pts[k] = make_float4(x, y, z, valid);
    }
}

// ---------------- kernel 2: WMMA distance-min ----------------
// Each wave handles one tile of 16 points; loops over all 553 vertex tiles.
// D[m,n] = A(16x4) x B(4x16) = -2 p.v + |v|^2  ; running min in the 8 D VGPRs.
__global__ __launch_bounds__(256)
void wmma_min_kernel(const float4* __restrict__ Pts, const float2* __restrict__ Bsw,
                     float* __restrict__ d2v) {
    __shared__ float2 ldsB[CHUNK_TILES * 32];   // 16 KB, already in lane order

    const int tid    = threadIdx.x;
    const int lane   = tid & 31;
    const int waveId = tid >> 5;
    const int b      = blockIdx.x >> 7;                       // 128 blocks per batch
    const int mtile  = (blockIdx.x & 127) * WAVES_PER_BLOCK + waveId;
    const int m0     = mtile * 16;

    // --- A matrix: 16x4 f32 -> 2 VGPRs (select once per wave).
    // lanes 0-15: (K0,K1) = (px,py) of point m0+lane ; lanes 16-31: (K2,K3) = (pz, valid)
    const float4 pv = Pts[b * NP + m0 + (lane & 15)];
    v2f a;
    a[0] = (lane < 16) ? pv.x : pv.z;
    a[1] = (lane < 16) ? pv.y : pv.w;

    float acc[8];
#pragma unroll
    for (int k = 0; k < 8; ++k) acc[k] = 3.0e38f;

    // stage via float4 (two lanes per copy)
    const float4* BswG4 = reinterpret_cast<const float4*>(Bsw);
    float4*       lds4  = reinterpret_cast<float4*>(ldsB);

    for (int ct = 0; ct < NTILES; ct += CHUNK_TILES) {
        const int tilesHere = min(CHUNK_TILES, NTILES - ct);
        __syncthreads();
        for (int v = tid; v < tilesHere * 16; v += 256)
            lds4[v] = BswG4[(b * NTILES + ct) * 16 + v];
        __syncthreads();

        for (int t = 0; t < tilesHere; ++t) {
            const float2 q = ldsB[t * 32 + lane];   // ds_load_b64, no swizzle needed
            v2f bm;
            bm[0] = q.x;
            bm[1] = q.y;
            v8f c = {};
            v8f d = __builtin_amdgcn_wmma_f32_16x16x4_f32(
                /*neg_a=*/false, a, /*neg_b=*/false, bm,
                /*c_mod=*/(short)0, c, /*reuse_a=*/false, /*reuse_b=*/false);
#pragma unroll
            for (int k = 0; k < 8; ++k) acc[k] = fminf(acc[k], d[k]);
        }
    }

    // --- reduce min over the 16 columns held across each 16-lane half.
#pragma unroll
    for (int j = 0; j < 8; ++j) {
        float r = acc[j];
        r = fminf(r, __shfl_xor(r, 1, 32));
        r = fminf(r, __shfl_xor(r, 2, 32));
        r = fminf(r, __shfl_xor(r, 4, 32));
        r = fminf(r, __shfl_xor(r, 8, 32));
        if ((lane & 15) == 0) {
            // lane 0 -> row j (M=j), lane 16 -> row j+8 (M=j+8)
            const int m = m0 + j + ((lane >> 4) << 3);
            const float4 p = Pts[b * NP + m];
            const float d2 = r + p.x * p.x + p.y * p.y + p.z * p.z;
            d2v[b * NP + m] = p.w * d2;   // p.w = valid flag (0 or 1)
        }
    }
}

// ---------------- kernel 3: deterministic final reduction ----------------
__global__ __launch_bounds__(256)
void reduce_kernel(const float* __restrict__ d2v, const float4* __restrict__ Pts,
                   float* __restrict__ out) {
    __shared__ float sd[NB][256];
    __shared__ float sc[NB][256];
    const int tid = threadIdx.x;
    for (int b = 0; b < NB; ++b) {
        float s = 0.0f, c = 0.0f;
        for (int n = tid; n < NP; n += 256) {
            s += d2v[b * NP + n];
            c += Pts[b * NP + n].w;
        }
        sd[b][tid] = s;
        sc[b][tid] = c;
    }
    __syncthreads();
    for (int off = 128; off > 0; off >>= 1) {
        if (tid < off) {
            for (int b = 0; b < NB; ++b) {
                sd[b][tid] += sd[b][tid + off];
                sc[b][tid] += sc[b][tid + off];
            }
        }
        __syncthreads();
    }
    if (tid == 0) {
        out[0] = 0.5f * (sd[0][0] / sc[0][0] + sd[1][0] / sc[1][0]);
    }
}

// ---------------- launcher ----------------
extern "C" void kernel_launch(void* const* d_in, const int* in_sizes, int n_in,
                              void* d_out, int out_size, void* d_ws, size_t ws_size,
                              hipStream_t stream) {
    (void)in_sizes; (void)n_in; (void)out_size; (void)ws_size;
    const float* verts = (const float*)d_in[0];   // (2,3,32,4,32) f32
    const float* pc    = (const float*)d_in[1];   // (2,3,16384) f32
    float* out = (float*)d_out;

    char* ws = (char*)d_ws;
    float2* Bsw = (float2*)ws;                                          // 2*553*32 float2
    float4* Pts = (float4*)(ws + (size_t)NB * NTILES * 32 * sizeof(float2));
    float*  d2v = (float*)(ws + (size_t)NB * NTILES * 32 * sizeof(float2)
                              + (size_t)NB * NP * sizeof(float4));

    const int prep_items = NB * NTILES * 32 + NB * NP;
    prep_kernel<<<(prep_items + 255) / 256, 256, 0, stream>>>(verts, pc, Bsw, Pts);
    wmma_min_kernel<<<NB * (MTILES / WAVES_PER_BLOCK), 256, 0, stream>>>(Pts, Bsw, d2v);
    reduce_kernel<<<1, 256, 0, stream>>>(d2v, Pts, out);
}